// GPPO_60404420051053
// MI455X (gfx1250) — compile-verified
//
#include <hip/hip_runtime.h>
#include <hip/hip_bf16.h>
#include <math.h>

typedef float v2f __attribute__((ext_vector_type(2)));
typedef float v8f __attribute__((ext_vector_type(8)));

// ---------------- layer-1 edge aggregation (D_IN = 2) ----------------
__global__ void gin_agg2_kernel(const int* __restrict__ src, const int* __restrict__ dst,
                                const float* __restrict__ x, float* __restrict__ agg, int E) {
    int e = blockIdx.x * blockDim.x + threadIdx.x;
    if (e >= E) return;
    int s = src[e], d = dst[e];
    atomicAdd(&agg[d * 2 + 0], x[s * 2 + 0]);
    atomicAdd(&agg[d * 2 + 1], x[s * 2 + 1]);
}

// ---------------- t1 = (x + agg1) @ W1 + b1   (2 -> 64) ----------------
__global__ void gin_lin1_kernel(const float* __restrict__ x, const float* __restrict__ agg1,
                                const float* __restrict__ W1, const float* __restrict__ b1,
                                float* __restrict__ T, int Nn) {
    long long t = (long long)blockIdx.x * blockDim.x + threadIdx.x;
    if (t >= (long long)Nn * 64) return;
    int n = (int)(t >> 6), j = (int)(t & 63);
    float z0 = x[n * 2 + 0] + agg1[n * 2 + 0];
    float z1 = x[n * 2 + 1] + agg1[n * 2 + 1];
    T[t] = fmaf(z0, W1[j], fmaf(z1, W1[64 + j], b1[j]));
}

// ---------------- per-column sum / sumsq for batchnorm ----------------
__global__ void gin_colstats_kernel(const float* __restrict__ T, float* __restrict__ stats,
                                    int rowsPerBlock, int Nn) {
    int j = threadIdx.x;              // 0..63
    int r0 = blockIdx.x * rowsPerBlock;
    float s = 0.f, q = 0.f;
    for (int i = 0; i < rowsPerBlock; ++i) {
        int r = r0 + i;
        if (r >= Nn) break;
        float v = T[(long long)r * 64 + j];
        s += v; q += v * v;
    }
    atomicAdd(&stats[j], s);
    atomicAdd(&stats[64 + j], q);
}

__global__ void gin_bnprep_kernel(const float* __restrict__ stats,
                                  const float* __restrict__ gamma, const float* __restrict__ beta,
                                  float* __restrict__ scale, float* __restrict__ shift, float invN) {
    int j = threadIdx.x;
    float mean = stats[j] * invN;
    float var  = stats[64 + j] * invN - mean * mean;
    float sc   = gamma[j] * rsqrtf(var + 1e-5f);
    scale[j] = sc;
    shift[j] = beta[j] - mean * sc;
}

// ---------------- WMMA fp32 GEMM: Out[N x 64] = act(A) @ W[64 x 64] + bias ----
// MODE 0: act(A) = A + A2 (elementwise)          (z = h + agg)
// MODE 1: act(A) = relu(A * scale + shift)       (post-batchnorm ReLU)
// Block = 128 threads (4 waves); wave w computes the 16x16 tile at cols w*16.
// 16 K-steps of V_WMMA_F32_16X16X4_F32 cover K = 64.
template <int MODE>
__global__ void gin_gemm64_kernel(const float* __restrict__ A, const float* __restrict__ A2,
                                  const float* __restrict__ scale, const float* __restrict__ shift,
                                  const float* __restrict__ W, const float* __restrict__ bias,
                                  float* __restrict__ Out, int Nn) {
    const int row0 = blockIdx.x * 16;
    const unsigned lane = threadIdx.x & 31u;
    const unsigned wave = threadIdx.x >> 5;
    const int col0 = (int)wave * 16;
    const int m    = (int)(lane & 15u);   // row-in-tile for A, col-in-tile for B/C
    const int half = (int)(lane >> 4);    // 0: K pair {0,1}, 1: K pair {2,3}

    v8f c = {};
    const int arow = (row0 + m < Nn) ? (row0 + m) : (Nn - 1);
    const long long abase = (long long)arow * 64;

#pragma unroll
    for (int kk = 0; kk < 16; ++kk) {
        const int ka = kk * 4 + half * 2;
        float ax, ay;
        if (MODE == 0) {
            ax = A[abase + ka]     + A2[abase + ka];
            ay = A[abase + ka + 1] + A2[abase + ka + 1];
        } else {
            float t0 = fmaf(A[abase + ka],     scale[ka],     shift[ka]);
            float t1 = fmaf(A[abase + ka + 1], scale[ka + 1], shift[ka + 1]);
            ax = t0 > 0.f ? t0 : 0.f;
            ay = t1 > 0.f ? t1 : 0.f;
        }
        v2f a; a.x = ax; a.y = ay;
        v2f b; b.x = W[(ka) * 64 + col0 + m];
               b.y = W[(ka + 1) * 64 + col0 + m];
        c = __builtin_amdgcn_wmma_f32_16x16x4_f32(false, a, false, b, (short)0, c, false, false);
    }

    const float bb = bias[col0 + m];
#pragma unroll
    for (int r = 0; r < 8; ++r) {
        int row = row0 + r + half * 8;
        if (row < Nn) Out[(long long)row * 64 + col0 + m] = c[r] + bb;
    }
}

// ---------------- layer-2 edge aggregation (64 channels) ----------------
__global__ void gin_agg64_kernel(const int* __restrict__ src, const int* __restrict__ dst,
                                 const float* __restrict__ H, float* __restrict__ G, long long total) {
    long long t = (long long)blockIdx.x * blockDim.x + threadIdx.x;
    if (t >= total) return;
    int e = (int)(t >> 6), cidx = (int)(t & 63);
    atomicAdd(&G[(long long)dst[e] * 64 + cidx], H[(long long)src[e] * 64 + cidx]);
}

// ---------------- contiguous mean pool ----------------
__global__ void gin_pool_kernel(const float* __restrict__ H, float* __restrict__ hg, int NPG) {
    int b = blockIdx.x, cidx = threadIdx.x;
    const float* base = H + (long long)b * NPG * 64 + cidx;
    float s = 0.f;
    for (int i = 0; i < NPG; ++i) s += base[(long long)i * 64];
    hg[b * 64 + cidx] = s / (float)NPG;
}

// ---------------- actor head: per-graph block, 64 threads = 64 candidates ----
__global__ void gin_actor_kernel(const float* __restrict__ H, const float* __restrict__ hg,
                                 const int* __restrict__ cand, const int* __restrict__ action,
                                 const float* __restrict__ aW1, const float* __restrict__ ab1,
                                 const float* __restrict__ aW2, const float* __restrict__ ab2,
                                 const float* __restrict__ aW3, const float* __restrict__ ab3,
                                 float* __restrict__ out) {
    __shared__ float feat[64][129];   // padded: stride%64 == 1 -> conflict-free
    __shared__ float zbuf[64][65];
    __shared__ float logits[64];

    const int b = blockIdx.x, k = threadIdx.x;   // k = candidate index
    const int node = cand[b * 64 + k];

    for (int cidx = 0; cidx < 64; ++cidx) feat[k][cidx]      = hg[b * 64 + cidx];
    for (int cidx = 0; cidx < 64; ++cidx) feat[k][64 + cidx] = H[(long long)node * 64 + cidx];

    // layer 1: 128 -> 64, tanh
    for (int j = 0; j < 64; ++j) {
        float acc = ab1[j];
        for (int i = 0; i < 128; ++i) acc = fmaf(feat[k][i], aW1[i * 64 + j], acc);
        zbuf[k][j] = tanhf(acc);
    }
    // layer 2: 64 -> 64, tanh, fused dot with aW3 (64 -> 1)
    float logit = ab3[0];
    for (int j = 0; j < 64; ++j) {
        float acc = ab2[j];
        for (int i = 0; i < 64; ++i) acc = fmaf(zbuf[k][i], aW2[i * 64 + j], acc);
        logit = fmaf(tanhf(acc), aW3[j], logit);
    }
    logits[k] = logit;
    __syncthreads();

    if (k == 0) {
        float mx = logits[0];
        for (int i = 1; i < 64; ++i) mx = fmaxf(mx, logits[i]);
        float se = 0.f;
        for (int i = 0; i < 64; ++i) se += expf(logits[i] - mx);
        float lse = mx + logf(se);
        int act = action[b];
        int aidx = 0;
        for (int i = 0; i < 64; ++i) { if (cand[b * 64 + i] == act) { aidx = i; break; } }
        out[b] = logits[aidx] - lse;
    }
}

extern "C" void kernel_launch(void* const* d_in, const int* in_sizes, int n_in,
                              void* d_out, int out_size, void* d_ws, size_t ws_size,
                              hipStream_t stream) {
    const float* x      = (const float*)d_in[0];
    const int*   edge   = (const int*)d_in[1];
    // d_in[2] = batch_ids (unused: groups are contiguous blocks of NPG nodes)
    const int*   cand   = (const int*)d_in[3];
    const int*   action = (const int*)d_in[4];
    const float* g1W1 = (const float*)d_in[5],  *g1b1 = (const float*)d_in[6];
    const float* g1g  = (const float*)d_in[7],  *g1be = (const float*)d_in[8];
    const float* g1W2 = (const float*)d_in[9],  *g1b2 = (const float*)d_in[10];
    const float* g2W1 = (const float*)d_in[11], *g2b1 = (const float*)d_in[12];
    const float* g2g  = (const float*)d_in[13], *g2be = (const float*)d_in[14];
    const float* g2W2 = (const float*)d_in[15], *g2b2 = (const float*)d_in[16];
    const float* aW1  = (const float*)d_in[17], *ab1  = (const float*)d_in[18];
    const float* aW2  = (const float*)d_in[19], *ab2  = (const float*)d_in[20];
    const float* aW3  = (const float*)d_in[21], *ab3  = (const float*)d_in[22];
    float* out = (float*)d_out;

    const int Nn  = in_sizes[0] / 2;
    const int E   = in_sizes[1] / 2;
    const int Bb  = in_sizes[4];
    const int NPG = Nn / Bb;

    const int* src = edge;
    const int* dst = edge + E;

    float* p = (float*)d_ws;
    float* agg1  = p; p += (size_t)Nn * 2;
    float* T     = p; p += (size_t)Nn * 64;
    float* H     = p; p += (size_t)Nn * 64;
    float* G     = p; p += (size_t)Nn * 64;
    float* stats = p; p += 128;
    float* scale = p; p += 64;
    float* shift = p; p += 64;
    float* hg    = p; p += (size_t)Bb * 64;

    const int rpb = (Nn + 255) / 256;
    const int gemmBlocks = (Nn + 15) / 16;

    // ---- layer 1 ----
    hipMemsetAsync(agg1, 0, (size_t)Nn * 2 * sizeof(float), stream);
    gin_agg2_kernel<<<(E + 255) / 256, 256, 0, stream>>>(src, dst, x, agg1, E);
    gin_lin1_kernel<<<(int)(((long long)Nn * 64 + 255) / 256), 256, 0, stream>>>(x, agg1, g1W1, g1b1, T, Nn);
    hipMemsetAsync(stats, 0, 128 * sizeof(float), stream);
    gin_colstats_kernel<<<256, 64, 0, stream>>>(T, stats, rpb, Nn);
    gin_bnprep_kernel<<<1, 64, 0, stream>>>(stats, g1g, g1be, scale, shift, 1.0f / (float)Nn);
    gin_gemm64_kernel<1><<<gemmBlocks, 128, 0, stream>>>(T, nullptr, scale, shift, g1W2, g1b2, H, Nn);

    // ---- layer 2 ----
    hipMemsetAsync(G, 0, (size_t)Nn * 64 * sizeof(float), stream);
    long long tot = (long long)E * 64;
    gin_agg64_kernel<<<(int)((tot + 255) / 256), 256, 0, stream>>>(src, dst, H, G, tot);
    gin_gemm64_kernel<0><<<gemmBlocks, 128, 0, stream>>>(H, G, nullptr, nullptr, g2W1, g2b1, T, Nn);
    hipMemsetAsync(stats, 0, 128 * sizeof(float), stream);
    gin_colstats_kernel<<<256, 64, 0, stream>>>(T, stats, rpb, Nn);
    gin_bnprep_kernel<<<1, 64, 0, stream>>>(stats, g2g, g2be, scale, shift, 1.0f / (float)Nn);
    gin_gemm64_kernel<1><<<gemmBlocks, 128, 0, stream>>>(T, nullptr, scale, shift, g2W2, g2b2, H, Nn);

    // ---- pool + actor ----
    gin_pool_kernel<<<Bb, 64, 0, stream>>>(H, hg, NPG);
    gin_actor_kernel<<<Bb, 64, 0, stream>>>(H, hg, cand, action,
                                            aW1, ab1, aW2, ab2, aW3, ab3, out);
}